// UpsampleLoss_38087769981579
// MI455X (gfx1250) — compile-verified
//
#include <hip/hip_runtime.h>
#include <hip/hip_bf16.h>

typedef __attribute__((ext_vector_type(2))) float v2f;
typedef __attribute__((ext_vector_type(8))) float v8f;

#define NPTS 4096
#define NB 4
#define NPOINT 204            // int(4096 * 0.05)
#define AUCTION_EPS_C 0.02f
#define H_REP_C 0.0005f

// ---------- CDNA5 async global->LDS copy (ASYNCcnt path) ----------
// SADDR form: per-lane LDS byte address in vdst VGPR, per-lane 32-bit byte
// offset in vaddr VGPR, uniform 64-bit base in SGPR pair.
__device__ __forceinline__ void async_g2l_b128(unsigned lds_addr,
                                               const void* gbase,
                                               unsigned goff) {
  asm volatile("global_load_async_to_lds_b128 %0, %1, %2"
               :: "v"(lds_addr), "v"(goff), "s"(gbase)
               : "memory");
}
__device__ __forceinline__ void wait_async0() {
  asm volatile("s_wait_asynccnt 0x0" ::: "memory");
}
// low 32 bits of a generic LDS pointer == LDS byte offset (ISA aperture rule)
__device__ __forceinline__ unsigned lds_off32(const void* p) {
  return (unsigned)(uintptr_t)p;
}

// ---------- helpers ----------
__device__ __forceinline__ void ins5(float* a, float v) {
  // keep a[0..4] the 5 smallest seen, sorted ascending
  if (v >= a[4]) return;
  a[4] = v;
#pragma unroll
  for (int k = 4; k > 0; --k) {
    if (a[k] < a[k - 1]) { float t = a[k]; a[k] = a[k - 1]; a[k - 1] = t; }
  }
}

__global__ void init_acc_kernel(float* acc) {
  if (threadIdx.x < 3) acc[threadIdx.x] = 0.0f;
}

// ---------- repulsion loss: WMMA pairwise-distance tiles ----------
// grid: (NPTS/16, NB); block: 256 threads (8 waves of 32)
// dynamic LDS: 16x2048 tile + staged points (48KB) + norms (16KB) + scratch
__global__ __launch_bounds__(256) void repulsion_wmma_kernel(
    const float* __restrict__ pred, float* __restrict__ rep_acc) {
  extern __shared__ char smem[];
  float* d2s   = (float*)smem;              // 16 x 2048 distances (one half)
  float* pts   = d2s + 16 * 2048;           // 4096 x 3 staged points (16B aligned)
  float* pnorm = pts + NPTS * 3;            // 4096 norms
  float* run   = pnorm + NPTS;              // 16 x 5 running smallest
  float* cand  = run + 16 * 5;              // 16 x 16 x 5 merge candidates

  const int b   = blockIdx.y;
  const int r0  = blockIdx.x * 16;
  const int tid = threadIdx.x;
  const float* P = pred + (size_t)b * NPTS * 3;

  // async-stage the whole batch's points into LDS (48KB, B128 = 12 iters/thread)
  {
    const unsigned nbytes = (unsigned)NPTS * 3u * 4u;  // 49152
    unsigned ldsb = lds_off32(pts);
    for (unsigned off = (unsigned)tid * 16u; off < nbytes; off += 256u * 16u)
      async_g2l_b128(ldsb + off, P, off);
  }
  if (tid < 16) {
#pragma unroll
    for (int k = 0; k < 5; ++k) run[tid * 5 + k] = 3.4e38f;
  }
  wait_async0();
  __syncthreads();

  // per-point squared norms from LDS
  for (int j = tid; j < NPTS; j += 256) {
    float x = pts[j * 3 + 0], y = pts[j * 3 + 1], z = pts[j * 3 + 2];
    pnorm[j] = x * x + y * y + z * z;
  }
  __syncthreads();

  const int  lane = tid & 31;
  const int  wave = tid >> 5;
  const int  m    = lane & 15;
  const bool hiL  = lane >= 16;

  // A (16x4 f32): lanes 0-15 hold K=0 (vgpr0), K=1 (vgpr1) for M=lane;
  //               lanes 16-31 hold K=2, K=3 for M=lane-16.
  // Fold the -2 of d2 = |a|^2+|b|^2-2ab into A; K=3 is a zero pad plane.
  float ax = pts[(r0 + m) * 3 + 0];
  float ay = pts[(r0 + m) * 3 + 1];
  float az = pts[(r0 + m) * 3 + 2];
  v2f A;
  A.x = -2.0f * (hiL ? az : ax);
  A.y = hiL ? 0.0f : (-2.0f * ay);

  for (int half = 0; half < 2; ++half) {
    const int c0 = half * 2048;
    // phase 1: each wave computes 16x16 WMMA tiles into LDS
    for (int t = wave; t < 128; t += 8) {
      const int col = c0 + t * 16 + m;
      float bx = pts[col * 3 + 0];
      float by = pts[col * 3 + 1];
      float bz = pts[col * 3 + 2];
      float nb = pnorm[col];
      // B (4x16 f32): lanes 0-15 hold rows K=0 (vgpr0), K=1 (vgpr1) for N=lane;
      //               lanes 16-31 hold rows K=2, K=3 for N=lane-16.
      v2f Bv;
      Bv.x = hiL ? bz : bx;
      Bv.y = hiL ? 0.0f : by;
      // C (16x16 f32, 8 VGPRs): vgpr r, lane l -> (M = r + 8*(l>=16), N = l%16)
      v8f C;
#pragma unroll
      for (int r = 0; r < 8; ++r) C[r] = pnorm[r0 + r + (hiL ? 8 : 0)] + nb;
      v8f D = __builtin_amdgcn_wmma_f32_16x16x4_f32(
          false, A, false, Bv, (short)0, C, false, false);
#pragma unroll
      for (int r = 0; r < 8; ++r) {
        float v = D[r];
        v = v > 0.0f ? v : 0.0f;   // jnp.maximum(d2, 0)
        d2s[(r + (hiL ? 8 : 0)) * 2048 + t * 16 + m] = v;
      }
    }
    __syncthreads();

    // phase 2: 16 threads per row scan strided columns, keep 5 smallest
    {
      const int row = tid >> 4;
      const int sub = tid & 15;
      float t5[5];
#pragma unroll
      for (int k = 0; k < 5; ++k) t5[k] = 3.4e38f;
      for (int j = sub; j < 2048; j += 16) ins5(t5, d2s[row * 2048 + j]);
#pragma unroll
      for (int k = 0; k < 5; ++k) cand[row * 80 + sub * 5 + k] = t5[k];
    }
    __syncthreads();
    if (tid < 16) {
      float m5[5];
#pragma unroll
      for (int k = 0; k < 5; ++k) m5[k] = run[tid * 5 + k];
      for (int q = 0; q < 80; ++q) ins5(m5, cand[tid * 80 + q]);
#pragma unroll
      for (int k = 0; k < 5; ++k) run[tid * 5 + k] = m5[k];
    }
    __syncthreads();
  }

  if (tid < 16) {
    // run[0] is self (d2 ~ 0); neighbors are run[1..4]
    float s = 0.0f;
#pragma unroll
    for (int k = 1; k < 5; ++k) {
      float v = H_REP_C - run[tid * 5 + k];
      s += v > 0.0f ? v : 0.0f;
    }
    atomicAdd(rep_acc, s);
  }
}

// ---------- EMD loss: Jacobi auction per batch ----------
// grid: NB; block: 1024; dynamic LDS: bid(u64)+gt xyz/norms+price+assign+owner
__global__ __launch_bounds__(1024) void emd_auction_kernel(
    const float* __restrict__ pred, const float* __restrict__ gt,
    const float* __restrict__ radius, float* __restrict__ emd_acc) {
  extern __shared__ char smem[];
  unsigned long long* bid = (unsigned long long*)smem;  // NPTS (16B aligned)
  float* gxyz = (float*)(bid + NPTS);                   // NPTS*3 (16B aligned)
  float* gn    = gxyz + NPTS * 3;
  float* price = gn + NPTS;
  int* assign = (int*)(price + NPTS);
  int* owner  = assign + NPTS;
  __shared__ int cnt;

  const int b   = blockIdx.x;
  const int tid = threadIdx.x;
  const float* G = gt + (size_t)b * NPTS * 3;
  const float* P = pred + (size_t)b * NPTS * 3;

  // async-stage gt into LDS while initializing auction state
  {
    const unsigned nbytes = (unsigned)NPTS * 3u * 4u;  // 49152
    unsigned ldsb = lds_off32(gxyz);
    for (unsigned off = (unsigned)tid * 16u; off < nbytes; off += 1024u * 16u)
      async_g2l_b128(ldsb + off, G, off);
  }
  for (int j = tid; j < NPTS; j += 1024) {
    price[j] = 0.0f; assign[j] = -1; owner[j] = -1;
  }
  wait_async0();
  __syncthreads();
  for (int j = tid; j < NPTS; j += 1024) {
    float x = gxyz[j * 3 + 0], y = gxyz[j * 3 + 1], z = gxyz[j * 3 + 2];
    gn[j] = x * x + y * y + z * z;
  }
  __syncthreads();

  for (int it = 0; it < 64; ++it) {
    if (tid == 0) cnt = 0;
    for (int j = tid; j < NPTS; j += 1024) bid[j] = 0ull;
    __syncthreads();
    int lu = 0;
    for (int i = tid; i < NPTS; i += 1024)
      if (assign[i] < 0) lu++;
    if (lu) atomicAdd(&cnt, lu);
    __syncthreads();
    if (cnt == 0) break;

    // bidding: each unassigned row scans all columns for top-2 value
    for (int i = tid; i < NPTS; i += 1024) {
      if (assign[i] >= 0) continue;
      float px = P[i * 3 + 0], py = P[i * 3 + 1], pz = P[i * 3 + 2];
      float pn = px * px + py * py + pz * pz;
      float b1 = -3.4e38f, b2 = -3.4e38f;
      int jb = 0;
      for (int j = 0; j < NPTS; ++j) {
        float d2 = pn + gn[j] -
                   2.0f * (px * gxyz[j * 3 + 0] + py * gxyz[j * 3 + 1] +
                           pz * gxyz[j * 3 + 2]);
        d2 = d2 > 0.0f ? d2 : 0.0f;
        float v = -d2 - price[j];
        if (v > b1) { b2 = b1; b1 = v; jb = j; }
        else if (v > b2) { b2 = v; }
      }
      float bidv = price[jb] + (b1 - b2) + AUCTION_EPS_C;  // > 0
      // pack: max bid wins, ties -> smallest row index
      unsigned long long key =
          ((unsigned long long)__float_as_uint(bidv) << 32) |
          (unsigned long long)(0xFFFFFFFFu - (unsigned)i);
      atomicMax(&bid[jb], key);
    }
    __syncthreads();

    // resolution
    for (int j = tid; j < NPTS; j += 1024) {
      unsigned long long k = bid[j];
      if (k != 0ull) {
        int w = (int)(0xFFFFFFFFu - (unsigned)(k & 0xFFFFFFFFull));
        float nbid = __uint_as_float((unsigned)(k >> 32));
        int prev = owner[j];
        if (prev >= 0) assign[prev] = -1;
        assign[w] = j;
        owner[j] = w;
        price[j] = nbid;
      }
    }
    __syncthreads();
  }

  // fallback (argmin d2 + price) + squared-error accumulation
  float rb = radius[b];
  float scale = 1.0f / (3.0f * (float)NPTS * (float)NB * rb);
  float local = 0.0f;
  for (int i = tid; i < NPTS; i += 1024) {
    float px = P[i * 3 + 0], py = P[i * 3 + 1], pz = P[i * 3 + 2];
    float pn = px * px + py * py + pz * pz;
    int j = assign[i];
    if (j < 0) {
      float best = 3.4e38f;
      int jb = 0;
      for (int jj = 0; jj < NPTS; ++jj) {
        float d2 = pn + gn[jj] -
                   2.0f * (px * gxyz[jj * 3 + 0] + py * gxyz[jj * 3 + 1] +
                           pz * gxyz[jj * 3 + 2]);
        d2 = d2 > 0.0f ? d2 : 0.0f;
        float v = d2 + price[jj];
        if (v < best) { best = v; jb = jj; }
      }
      j = jb;
    }
    float dx = px - gxyz[j * 3 + 0];
    float dy = py - gxyz[j * 3 + 1];
    float dz = pz - gxyz[j * 3 + 2];
    local += dx * dx + dy * dy + dz * dz;
  }
  atomicAdd(emd_acc, local * scale);
}

// ---------- FPS (start at index 0), writes seed coordinates ----------
__global__ __launch_bounds__(256) void fps_kernel(
    const float* __restrict__ pred, float* __restrict__ seeds) {
  __shared__ float dists[NPTS];
  __shared__ float rmax[256];
  __shared__ int   ridx[256];
  __shared__ int   far_s;
  const int b   = blockIdx.x;
  const int tid = threadIdx.x;
  const float* P = pred + (size_t)b * NPTS * 3;
  for (int i = tid; i < NPTS; i += 256) dists[i] = 3.4e38f;
  if (tid == 0) far_s = 0;
  __syncthreads();
  for (int s = 0; s < NPOINT; ++s) {
    int far = far_s;
    float fx = P[far * 3 + 0], fy = P[far * 3 + 1], fz = P[far * 3 + 2];
    if (tid == 0) {
      seeds[((size_t)b * NPOINT + s) * 3 + 0] = fx;
      seeds[((size_t)b * NPOINT + s) * 3 + 1] = fy;
      seeds[((size_t)b * NPOINT + s) * 3 + 2] = fz;
    }
    float lm = -1.0f;
    int   li = 0;
    for (int i = tid; i < NPTS; i += 256) {
      float dx = P[i * 3 + 0] - fx, dy = P[i * 3 + 1] - fy, dz = P[i * 3 + 2] - fz;
      float d  = dx * dx + dy * dy + dz * dz;
      float nd = fminf(dists[i], d);
      dists[i] = nd;
      if (nd > lm) { lm = nd; li = i; }
    }
    rmax[tid] = lm; ridx[tid] = li;
    __syncthreads();
    for (int off = 128; off > 0; off >>= 1) {
      if (tid < off) {
        if (rmax[tid + off] > rmax[tid] ||
            (rmax[tid + off] == rmax[tid] && ridx[tid + off] < ridx[tid])) {
          rmax[tid] = rmax[tid + off];
          ridx[tid] = ridx[tid + off];
        }
      }
      __syncthreads();
    }
    if (tid == 0) far_s = ridx[0];
    __syncthreads();
  }
}

// ---------- uniform loss: ordered ball-query + in-group NN ----------
// grid: NB*NPOINT; block: 128
__global__ __launch_bounds__(128) void uniform_kernel(
    const float* __restrict__ pred, const float* __restrict__ seeds,
    float* __restrict__ uni_acc) {
  __shared__ int   cnts[128];
  __shared__ int   offs[128];
  __shared__ int   cand[64];
  __shared__ float mx[64], my[64], mz[64];
  __shared__ int   total_s;
  __shared__ float gsum;
  const int g = blockIdx.x;
  const int b = g / NPOINT;
  const int t = threadIdx.x;
  const float sx = seeds[g * 3 + 0];
  const float sy = seeds[g * 3 + 1];
  const float sz = seeds[g * 3 + 2];
  const float* P = pred + (size_t)b * NPTS * 3;
  const float expect = sqrtf(3.14159265358979f / (float)NPTS);
  const float ps[5]  = {0.004f, 0.006f, 0.008f, 0.01f, 0.012f};
  const int   nss[5] = {16, 24, 32, 40, 49};

  for (int pi = 0; pi < 5; ++pi) {
    const float r2 = ps[pi];  // r^2 = p * UNI_RADIUS^2
    const int   ns = nss[pi];
    const int   i0 = t * 32;
    int c = 0;
    for (int i = i0; i < i0 + 32; ++i) {
      float dx = P[i * 3 + 0] - sx, dy = P[i * 3 + 1] - sy, dz = P[i * 3 + 2] - sz;
      if (dx * dx + dy * dy + dz * dz < r2) c++;
    }
    cnts[t] = c;
    __syncthreads();
    if (t == 0) {
      int run = 0;
      for (int k = 0; k < 128; ++k) { offs[k] = run; run += cnts[k]; }
      total_s = run;
      gsum = 0.0f;
    }
    __syncthreads();
    int pos = offs[t];
    if (pos < 64) {
      for (int i = i0; i < i0 + 32 && pos < 64; ++i) {
        float dx = P[i * 3 + 0] - sx, dy = P[i * 3 + 1] - sy, dz = P[i * 3 + 2] - sz;
        if (dx * dx + dy * dy + dz * dz < r2) cand[pos++] = i;
      }
    }
    __syncthreads();
    const int total = total_s;
    if (t < ns) {
      int id = (t < total) ? cand[t] : cand[0];  // pad with first member
      mx[t] = P[id * 3 + 0];
      my[t] = P[id * 3 + 1];
      mz[t] = P[id * 3 + 2];
    }
    __syncthreads();
    if (t < ns) {
      float b1 = 3.4e38f, b2 = 3.4e38f;  // two smallest incl. self(0)
      for (int k = 0; k < ns; ++k) {
        float dx = mx[t] - mx[k], dy = my[t] - my[k], dz = mz[t] - mz[k];
        float d2 = dx * dx + dy * dy + dz * dz;
        if (d2 < b1) { b2 = b1; b1 = d2; }
        else if (d2 < b2) { b2 = d2; }
      }
      float dist = sqrtf(fmaxf(b2, 1e-12f));
      atomicAdd(&gsum, fabsf(dist + 1e-8f));
    }
    __syncthreads();
    if (t == 0) {
      float uni  = gsum / (float)ns;
      float d    = uni - expect;
      float term = d * d / (expect + 1e-8f);
      float pw   = ps[pi] * 100.0f;
      atomicAdd(uni_acc,
                term * pw * pw / (5.0f * (float)NB * (float)NPOINT));
    }
    __syncthreads();
  }
}

// ---------- finalize ----------
__global__ void finalize_kernel(const float* __restrict__ acc,
                                float* __restrict__ out) {
  out[0] = 100.0f * acc[0];
  out[1] = 10.0f * acc[1];
  out[2] = 5.0f * acc[2] / ((float)NB * (float)NPTS * 4.0f);
}

extern "C" void kernel_launch(void* const* d_in, const int* in_sizes, int n_in,
                              void* d_out, int out_size, void* d_ws, size_t ws_size,
                              hipStream_t stream) {
  const float* pred   = (const float*)d_in[0];
  const float* gt     = (const float*)d_in[1];
  const float* radius = (const float*)d_in[2];

  float* acc   = (float*)d_ws;   // acc[0]=emd, acc[1]=uniform, acc[2]=repulsion
  float* seeds = acc + 8;        // NB*NPOINT*3 floats

  init_acc_kernel<<<1, 32, 0, stream>>>(acc);

  // EMD auction: LDS = bid(u64)*N + gxyz(3N) + gn/price(N each) + assign/owner
  size_t emd_lds = (size_t)NPTS * 8 + (size_t)NPTS * 4 * 7;  // 147456 B
  emd_auction_kernel<<<NB, 1024, emd_lds, stream>>>(pred, gt, radius, acc + 0);

  fps_kernel<<<NB, 256, 0, stream>>>(pred, seeds);
  uniform_kernel<<<NB * NPOINT, 128, 0, stream>>>(pred, seeds, acc + 1);

  // Repulsion: 16x2048 tile + staged pts (3N) + norms (N) + 80 run + 1280 cand
  size_t rep_lds =
      (size_t)(16 * 2048 + NPTS * 3 + NPTS + 80 + 1280) * sizeof(float);  // 202048 B
  dim3 rep_grid(NPTS / 16, NB);
  repulsion_wmma_kernel<<<rep_grid, 256, rep_lds, stream>>>(pred, acc + 2);

  finalize_kernel<<<1, 1, 0, stream>>>(acc, (float*)d_out);
}